// Peak2Peak_38955353375392
// MI455X (gfx1250) — compile-verified
//
#include <hip/hip_runtime.h>

typedef __attribute__((ext_vector_type(4))) unsigned int u32x4;
typedef __attribute__((ext_vector_type(8))) int          i32x8;
typedef __attribute__((ext_vector_type(4))) int          i32x4;
typedef __attribute__((ext_vector_type(4))) float        f32x4;

#define BATCH   32
#define SEQ     16384
#define FEAT    8
#define KSZ     9
#define NOUT    (SEQ - KSZ)          /* 16375 */
#define TILE_N  1024
#define HALO    7
#define ROWS    (TILE_N + HALO)      /* 1031 input rows needed per tile */
#define NTILES  ((NOUT + TILE_N - 1) / TILE_N)   /* 16 */

__global__ __launch_bounds__(256)
void Peak2Peak_38955353375392_kernel(const float* __restrict__ x,
                                     const float* __restrict__ W,
                                     const float* __restrict__ bias,
                                     float* __restrict__ out) {
    __shared__ float tile[ROWS * FEAT];   // 32,992 bytes

    const int tileIdx = blockIdx.x;
    const int b       = blockIdx.y;
    const int n0      = tileIdx * TILE_N;

    // ---- TDM: async DMA rows [n0, n0+ROWS) of batch b from HBM into LDS ----
    // Issued by wave 0 only (TDM issue ignores EXEC, so guard at wave granularity).
    if (threadIdx.x < 32) {
        unsigned long long startElem  = ((unsigned long long)b * SEQ + (unsigned long long)n0) * FEAT;
        unsigned long long gaddr      = (unsigned long long)(uintptr_t)x + startElem * 4ull;
        unsigned long long totalElems = (unsigned long long)BATCH * SEQ * FEAT;
        unsigned int remain    = (unsigned int)(totalElems - startElem); // OOB clamp for the very last tile
        unsigned int tileElems = ROWS * FEAT;                            // 8248 elements (1-D transfer)
        unsigned int ldsOff    = (unsigned int)(uintptr_t)(&tile[0]);    // low 32 bits of flat LDS addr

        // D# group 0 (ISA 8.3): count=1 | lds_addr | global_addr[56:0] | type=2
        u32x4 g0;
        g0[0] = 1u;
        g0[1] = ldsOff;
        g0[2] = (unsigned int)(gaddr & 0xFFFFFFFFull);
        g0[3] = ((unsigned int)(gaddr >> 32) & 0x01FFFFFFu) | (2u << 30);

        // D# group 1 (ISA 8.4): wg_mask=0, data_size=2 (4B), no pad/iterate,
        // tensor_dim0=remain (OOB->0), tensor_dim1=1, tile_dim0=tileElems, tile_dim1/2=0
        i32x8 g1;
        g1[0] = (int)(2u << 16);                        // [17:16] data_size = 4 bytes
        g1[1] = (int)((remain & 0xFFFFu) << 16);        // tensor_dim0[15:0] @ bits 63:48
        g1[2] = (int)((remain >> 16) | (1u << 16));     // tensor_dim0[31:16] | tensor_dim1[15:0]=1
        g1[3] = (int)((tileElems & 0xFFFFu) << 16);     // tile_dim0 @ bits 127:112
        g1[4] = 0;                                      // tile_dim1=0 (unused), tile_dim2=0
        g1[5] = (int)tileElems;                         // tensor_dim0_stride lo (unused for 1-D)
        g1[6] = 0;                                      // stride hi / tensor_dim1_stride lo
        g1[7] = 0;                                      // tensor_dim1_stride hi

        i32x4 gz4 = {0, 0, 0, 0};                       // groups 2/3 unused (<=2D tensor)
        i32x8 gz8 = {0, 0, 0, 0, 0, 0, 0, 0};           // extended words (unused)
        __builtin_amdgcn_tensor_load_to_lds(g0, g1, gz4, gz4, gz8, 0);
    }

    // ---- Uniform weight/bias loads overlap with the tensor DMA ----
    float w0[FEAT];              // W[f][0]
    float wk[KSZ - 1][FEAT];     // wk[k][f] = W[f][k+1]
    #pragma unroll
    for (int f = 0; f < FEAT; ++f) {
        w0[f] = W[f * KSZ];
        #pragma unroll
        for (int k = 0; k < KSZ - 1; ++k)
            wk[k][f] = W[f * KSZ + k + 1];
    }
    const float b0 = bias[0];

    if (threadIdx.x < 32)
        __builtin_amdgcn_s_wait_tensorcnt(0);
    __syncthreads();

    const f32x4* tf4 = (const f32x4*)tile;

    // Each thread computes 4 rows, strided by 256 -> fully coalesced stores.
    #pragma unroll
    for (int i = 0; i < TILE_N / 256; ++i) {
        const int l = (int)threadIdx.x + i * 256;
        const int n = n0 + l;
        if (n < NOUT) {
            float r0[FEAT];
            *(f32x4*)&r0[0] = tf4[l * 2 + 0];      // ds_load_b128
            *(f32x4*)&r0[4] = tf4[l * 2 + 1];

            float base[FEAT], acc[FEAT];
            #pragma unroll
            for (int f = 0; f < FEAT; ++f) {
                base[f] = r0[f] * w0[f];
                acc[f]  = 0.0f;
            }

            #pragma unroll
            for (int k = 0; k < KSZ - 1; ++k) {
                float rk[FEAT];
                *(f32x4*)&rk[0] = tf4[(l + k) * 2 + 0];
                *(f32x4*)&rk[4] = tf4[(l + k) * 2 + 1];
                #pragma unroll
                for (int f = 0; f < FEAT; ++f)
                    acc[f] += __builtin_fabsf(base[f] - rk[f] * wk[k][f]);
            }

            f32x4 o0, o1;
            o0.x = acc[0] + b0; o0.y = acc[1] + b0; o0.z = acc[2] + b0; o0.w = acc[3] + b0;
            o1.x = acc[4] + b0; o1.y = acc[5] + b0; o1.z = acc[6] + b0; o1.w = acc[7] + b0;

            size_t off = ((size_t)b * NOUT + (size_t)n) * FEAT;   // 32B-aligned
            __builtin_nontemporal_store(o0, (f32x4*)(out + off));       // streaming b128 store
            __builtin_nontemporal_store(o1, (f32x4*)(out + off + 4));
        }
    }
}

extern "C" void kernel_launch(void* const* d_in, const int* in_sizes, int n_in,
                              void* d_out, int out_size, void* d_ws, size_t ws_size,
                              hipStream_t stream) {
    (void)in_sizes; (void)n_in; (void)d_ws; (void)ws_size; (void)out_size;
    const float* x    = (const float*)d_in[0];
    const float* W    = (const float*)d_in[1];
    const float* bias = (const float*)d_in[2];
    float*       out  = (float*)d_out;

    dim3 grid(NTILES, BATCH);   // 16 x 32 = 512 workgroups
    dim3 block(256);            // 8 wave32 waves
    Peak2Peak_38955353375392_kernel<<<grid, block, 0, stream>>>(x, W, bias, out);
}